// GCNLayer_70420283785883
// MI455X (gfx1250) — compile-verified
//
#include <hip/hip_runtime.h>
#include <hip/hip_bf16.h>

typedef __attribute__((ext_vector_type(2))) float v2f;
typedef __attribute__((ext_vector_type(4))) float v4f;
typedef __attribute__((ext_vector_type(8))) float v8f;

#define IN_FEATS 256
#define OUT_FEATS 256

// ---------------------------------------------------------------------------
// Kernel 0: zero the aggregation workspace (scatter-add target).
// ---------------------------------------------------------------------------
__global__ __launch_bounds__(256) void gcn_zero_kernel(v4f* __restrict__ agg,
                                                       long n4) {
  long stride = (long)gridDim.x * blockDim.x;
  for (long i = (long)blockIdx.x * blockDim.x + threadIdx.x; i < n4; i += stride) {
    agg[i] = (v4f){0.f, 0.f, 0.f, 0.f};
  }
}

// ---------------------------------------------------------------------------
// Kernel 1: per-edge gather of RAW input features + scale + scatter-add.
//   agg[dst[e]] += h[src[e]] * sym_norm[e]
// (Linearity: segment_sum((h@W)[src]*n) == segment_sum(h[src]*n) @ W, so we
//  aggregate first and run the dense transform once afterwards.)
// 64 threads (2 waves) per edge, float4 per thread (256 feats).
// src/dst/norm are uniform within each wave -> scalarized loads.
// h (102 MB) and agg (102 MB) both fit in the 192 MB L2.
// ---------------------------------------------------------------------------
__global__ __launch_bounds__(256) void gcn_edge_scatter_kernel(
    const float* __restrict__ h, const float* __restrict__ sym_norm,
    const int* __restrict__ src, const int* __restrict__ dst,
    float* __restrict__ agg, int n_edges) {
  int t = blockIdx.x * 256 + threadIdx.x;
  int e = t >> 6;  // 64 threads per edge
  if (e >= n_edges) return;
  int f4 = t & 63;  // which float4 of the 256-float row

  const int   s  = src[e];
  const int   d  = dst[e];
  const float nv = sym_norm[e];

  const v4f* __restrict__ srow = (const v4f*)(h + (size_t)s * IN_FEATS);
  v4f val = srow[f4];
  float* __restrict__ op = agg + (size_t)d * IN_FEATS + (size_t)f4 * 4;

  unsafeAtomicAdd(op + 0, val.x * nv);
  unsafeAtomicAdd(op + 1, val.y * nv);
  unsafeAtomicAdd(op + 2, val.z * nv);
  unsafeAtomicAdd(op + 3, val.w * nv);
}

// ---------------------------------------------------------------------------
// Kernel 2: out = relu(agg @ W + b) using V_WMMA_F32_16X16X4_F32.
// Block = 128 threads (4 waves). Block handles 32 rows x 256 cols.
// Wave w handles N range [w*64, w*64+64) for BOTH 16-row M-tiles, so each
// B fragment is reused by two WMMAs (halves W traffic from L2 vs 16-row
// blocks). 8 x v8f accumulators = 64 VGPRs.
//
// Fragment layouts per CDNA5 ISA 7.12.2 (wave32):
//   A 16x4 :  lane L -> m = L%16 ; VGPR v -> k = kb + 2*(L/16) + v
//   B 4x16 :  lane L -> n = L%16 ; VGPR v -> k = kb + 2*(L/16) + v
//   C 16x16:  lane L -> n = L%16 ; VGPR v -> m = v + 8*(L/16)
// Bias + ReLU fused in the epilogue.
// ---------------------------------------------------------------------------
__global__ __launch_bounds__(128) void gcn_gemm_wmma_kernel(
    const float* __restrict__ agg, const float* __restrict__ W,
    const float* __restrict__ bias, float* __restrict__ out, int n_rows) {
  const int lane = threadIdx.x & 31;
  const int wave = threadIdx.x >> 5;  // 0..3
  const int row0 = blockIdx.x * 32;
  if (row0 >= n_rows) return;

  const int m     = lane & 15;
  const int khalf = (lane >> 4) << 1;  // 0 or 2
  const int nlane = wave * 64 + (lane & 15);

  const float* __restrict__ arow0 = agg + (size_t)(row0 + m) * IN_FEATS;
  const float* __restrict__ arow1 = arow0 + (size_t)16 * IN_FEATS;

  v8f acc00 = {}, acc01 = {}, acc02 = {}, acc03 = {};  // M-tile 0, N-tiles 0..3
  v8f acc10 = {}, acc11 = {}, acc12 = {}, acc13 = {};  // M-tile 1, N-tiles 0..3

  for (int kb = 0; kb < IN_FEATS; kb += 4) {
    const int k0 = kb + khalf;
    v2f a0, a1;
    a0.x = arow0[k0]; a0.y = arow0[k0 + 1];
    a1.x = arow1[k0]; a1.y = arow1[k0 + 1];

    const float* __restrict__ wk0 = W + (size_t)k0 * OUT_FEATS;
    const float* __restrict__ wk1 = wk0 + OUT_FEATS;
    v2f b0, b1, b2, b3;
    b0.x = wk0[nlane +  0]; b0.y = wk1[nlane +  0];
    b1.x = wk0[nlane + 16]; b1.y = wk1[nlane + 16];
    b2.x = wk0[nlane + 32]; b2.y = wk1[nlane + 32];
    b3.x = wk0[nlane + 48]; b3.y = wk1[nlane + 48];

    acc00 = __builtin_amdgcn_wmma_f32_16x16x4_f32(false, a0, false, b0, (short)0, acc00, false, false);
    acc10 = __builtin_amdgcn_wmma_f32_16x16x4_f32(false, a1, false, b0, (short)0, acc10, false, false);
    acc01 = __builtin_amdgcn_wmma_f32_16x16x4_f32(false, a0, false, b1, (short)0, acc01, false, false);
    acc11 = __builtin_amdgcn_wmma_f32_16x16x4_f32(false, a1, false, b1, (short)0, acc11, false, false);
    acc02 = __builtin_amdgcn_wmma_f32_16x16x4_f32(false, a0, false, b2, (short)0, acc02, false, false);
    acc12 = __builtin_amdgcn_wmma_f32_16x16x4_f32(false, a1, false, b2, (short)0, acc12, false, false);
    acc03 = __builtin_amdgcn_wmma_f32_16x16x4_f32(false, a0, false, b3, (short)0, acc03, false, false);
    acc13 = __builtin_amdgcn_wmma_f32_16x16x4_f32(false, a1, false, b3, (short)0, acc13, false, false);
  }

  // Fused bias + ReLU epilogue. Bias is per output column.
  const float bv0 = bias[nlane +  0];
  const float bv1 = bias[nlane + 16];
  const float bv2 = bias[nlane + 32];
  const float bv3 = bias[nlane + 48];

  const int mbase = (lane >> 4) * 8;  // lanes 16-31 hold M=8..15
#pragma unroll
  for (int v = 0; v < 8; ++v) {
    float* __restrict__ orow0 = out + (size_t)(row0 + mbase + v) * OUT_FEATS;
    float* __restrict__ orow1 = orow0 + (size_t)16 * OUT_FEATS;
    orow0[nlane +  0] = fmaxf(acc00[v] + bv0, 0.f);
    orow0[nlane + 16] = fmaxf(acc01[v] + bv1, 0.f);
    orow0[nlane + 32] = fmaxf(acc02[v] + bv2, 0.f);
    orow0[nlane + 48] = fmaxf(acc03[v] + bv3, 0.f);
    orow1[nlane +  0] = fmaxf(acc10[v] + bv0, 0.f);
    orow1[nlane + 16] = fmaxf(acc11[v] + bv1, 0.f);
    orow1[nlane + 32] = fmaxf(acc12[v] + bv2, 0.f);
    orow1[nlane + 48] = fmaxf(acc13[v] + bv3, 0.f);
  }
}

extern "C" void kernel_launch(void* const* d_in, const int* in_sizes, int n_in,
                              void* d_out, int out_size, void* d_ws, size_t ws_size,
                              hipStream_t stream) {
  const float* h        = (const float*)d_in[0];
  const float* sym_norm = (const float*)d_in[1];
  const float* W        = (const float*)d_in[2];
  const float* bias     = (const float*)d_in[3];
  const int*   src      = (const int*)d_in[4];
  const int*   dst      = (const int*)d_in[5];
  float* out = (float*)d_out;
  float* agg = (float*)d_ws;  // 100000 * 256 * 4 B = 102.4 MB scratch

  const int n_nodes = in_sizes[0] / IN_FEATS;  // 100000
  const int n_edges = in_sizes[1];             // 1600000

  const long n4 = (long)n_nodes * (IN_FEATS / 4);

  // 1) zero the aggregation buffer (harness poisons d_ws with 0xAA)
  gcn_zero_kernel<<<2048, 256, 0, stream>>>((v4f*)agg, n4);

  // 2) edge gather * norm on raw features, scatter-add onto dst rows
  {
    long total_threads = (long)n_edges * 64;
    int blocks = (int)((total_threads + 255) / 256);
    gcn_edge_scatter_kernel<<<blocks, 256, 0, stream>>>(h, sym_norm, src, dst, agg, n_edges);
  }

  // 3) out = relu(agg @ W + b)  (WMMA f32, fused epilogue).
  //    n_nodes is a multiple of 32 (100000 = 3125 * 32).
  gcn_gemm_wmma_kernel<<<(n_nodes + 31) / 32, 128, 0, stream>>>(agg, W, bias, out, n_nodes);
}